// RoiPairer_88313117540565
// MI455X (gfx1250) — compile-verified
//
#include <hip/hip_runtime.h>

// Problem constants (from reference)
#define Bk    4
#define Nk    24
#define Ck    256
#define Hk    100
#define Wk    100
#define SCALEk 0.25f
#define OUTk  7
#define Pk    276            // N*(N-1)/2
#define Mk    300            // N + P
#define GSk   14             // OUT*SR sample positions per axis
#define Kk    98             // folded K: 14 y-samples x 7 x-bins
#define KPADk 100            // K padded to multiple of 4
#define LDk   65             // LDS row stride (floats), padded vs 64
#define BINSk 49
#define CCHk  64             // channels per chunk (4 chunks)
#define TPBk  256

typedef float v2f __attribute__((ext_vector_type(2)));
typedef float v8f __attribute__((ext_vector_type(8)));

__global__ __launch_bounds__(TPBk) void roi_pair_wmma_kernel(
    const float* __restrict__ feat,    // [B, C, H, W]
    const float* __restrict__ boxes,   // [B, N, 4]
    float* __restrict__ out)           // [B*P, 3, C, 7, 7]
{
    const int roi   = blockIdx.x;          // 0..1199
    const int cbase = blockIdx.y * CCHk;   // 0,64,128,192
    const int b     = roi / Mk;
    const int m     = roi - b * Mk;
    const int tid   = threadIdx.x;
    const int lane  = tid & 31;
    const int wave  = tid >> 5;

    __shared__ float sT[KPADk * LDk];    // x-folded samples  [k][c]   26,000 B
    __shared__ float sPm[KPADk * LDk];   // pooling matrix    [k][bin] 26,000 B
    __shared__ float sP[CCHk * BINSk];   // pooled chunk      [c][bin] 12,544 B
    __shared__ int   sy0[GSk], sy1[GSk], sx0[GSk], sx1[GSk];
    __shared__ float sly[GSk], shy[GSk], svy[GSk];
    __shared__ float slx[GSk], shx[GSk], svx[GSk];

    // ---- RoI box (uniform per block) ----
    float x1, y1, x2, y2;
    if (m < Nk) {
        const float* bx = boxes + ((size_t)b * Nk + m) * 4;
        x1 = bx[0]; y1 = bx[1]; x2 = bx[2]; y2 = bx[3];
    } else {
        int pi = m - Nk;
        int i = 0;
        while (pi >= Nk - 1 - i) { pi -= Nk - 1 - i; ++i; }
        int j = i + 1 + pi;
        const float* bi = boxes + ((size_t)b * Nk + i) * 4;
        const float* bj = boxes + ((size_t)b * Nk + j) * 4;
        x1 = fminf(bi[0], bj[0]); y1 = fminf(bi[1], bj[1]);
        x2 = fmaxf(bi[2], bj[2]); y2 = fmaxf(bi[3], bj[3]);
    }
    x1 *= SCALEk; y1 *= SCALEk; x2 *= SCALEk; y2 *= SCALEk;
    const float bin_w = fmaxf(x2 - x1, 1.0f) * (1.0f / OUTk);
    const float bin_h = fmaxf(y2 - y1, 1.0f) * (1.0f / OUTk);

    // ---- sample taps (14 y, 14 x) ----
    if (tid < 2 * GSk) {
        int g = (tid < GSk) ? tid : tid - GSk;
        float grid = (float)(g >> 1) + ((float)(g & 1) + 0.5f) * 0.5f;
        if (tid < GSk) {
            float y = y1 + grid * bin_h;
            svy[g] = (y > -1.0f && y < (float)Hk) ? 1.0f : 0.0f;
            float yc = fminf(fmaxf(y, 0.0f), (float)(Hk - 1));
            float y0f = floorf(yc);
            int   y0  = (int)y0f;
            int   yp  = y0 + 1; if (yp > Hk - 1) yp = Hk - 1;
            sy0[g] = y0; sy1[g] = yp;
            float ly = yc - y0f;
            sly[g] = ly; shy[g] = 1.0f - ly;
        } else {
            float x = x1 + grid * bin_w;
            svx[g] = (x > -1.0f && x < (float)Wk) ? 1.0f : 0.0f;
            float xc = fminf(fmaxf(x, 0.0f), (float)(Wk - 1));
            float x0f = floorf(xc);
            int   x0  = (int)x0f;
            int   xp  = x0 + 1; if (xp > Wk - 1) xp = Wk - 1;
            sx0[g] = x0; sx1[g] = xp;
            float lx = xc - x0f;
            slx[g] = lx; shx[g] = 1.0f - lx;
        }
    }
    // zero K-pad rows of sT (rows 98,99 incl. pad column)
    if (tid < (KPADk - Kk) * LDk) sT[Kk * LDk + tid] = 0.0f;
    __syncthreads();

    // ---- pooling matrix P[k][bin]: branch-free build, covers pads ----
    for (int i = tid; i < KPADk * LDk; i += TPBk) {
        int k   = i / LDk;
        int col = i - k * LDk;
        float v = 0.0f;
        if (k < Kk && col < BINSk) {
            int yi = k / OUTk, pw = k - yi * OUTk;       // y-sample, x-bin
            int ph = col / OUTk, pb = col - ph * OUTk;   // bin row/col
            if ((yi >> 1) == ph && pw == pb) v = 0.25f * svy[yi];
        }
        sPm[i] = v;
    }

    // ---- gather: fold 2 x-subsamples per bin column -> sT[k][c] ----
    const size_t fbase = ((size_t)b * Ck + cbase) * (size_t)(Hk * Wk);
    for (int i = tid; i < Kk * CCHk; i += TPBk) {
        int k = i % Kk;            // lanes walk sample positions (x-contiguous)
        int c = i / Kk;
        int yi = k / OUTk, pw = k - (k / OUTk) * OUTk;
        int xa = 2 * pw, xb = xa + 1;
        const float* fp = feat + fbase + (size_t)c * (Hk * Wk);
        int r0 = sy0[yi] * Wk, r1 = sy1[yi] * Wk;
        float hy = shy[yi], ly = sly[yi];
        // sub-sample A
        int c0 = sx0[xa], c1 = sx1[xa];
        float va = shx[xa] * (hy * fp[r0 + c0] + ly * fp[r1 + c0])
                 + slx[xa] * (hy * fp[r0 + c1] + ly * fp[r1 + c1]);
        // sub-sample B
        int d0 = sx0[xb], d1 = sx1[xb];
        float vb = shx[xb] * (hy * fp[r0 + d0] + ly * fp[r1 + d0])
                 + slx[xb] * (hy * fp[r0 + d1] + ly * fp[r1 + d1]);
        sT[k * LDk + c] = svx[xa] * va + svx[xb] * vb;
    }
    __syncthreads();

    // ---- GEMM: D[c,bin] = sum_k sT[k][c] * P[k][bin]  (M=64, N=64, K=100) ----
    // Each wave owns bin-tile nT = (wave&3)*16 and channel tiles mT, mT+32
    // (its two 16x16 tiles share the B fragment -> load B once, 2 WMMAs).
    // V_WMMA_F32_16X16X4_F32 frag layout (wave32, 32-bit):
    //   A/B VGPR0: lanes0-15 K=k0, lanes16-31 K=k0+2; VGPR1: K=k0+1 / k0+3
    //   C/D vgpr r: lanes0-15 M=r, lanes16-31 M=r+8; N=lane%16
    const int half = lane >> 4;
    const int l15  = lane & 15;
    const int mT   = (wave >> 2) * 16;     // 0 or 16 (+32 for second tile)
    const int nT   = (wave & 3) * 16;      // bin tile
    {
        v8f acc0 = {0.f, 0.f, 0.f, 0.f, 0.f, 0.f, 0.f, 0.f};
        v8f acc1 = {0.f, 0.f, 0.f, 0.f, 0.f, 0.f, 0.f, 0.f};
        for (int k0 = 0; k0 < KPADk; k0 += 4) {
            const int ka = k0 + half * 2;
            v2f a0, a1, bb;
            a0.x = sT[ka * LDk + mT + l15];
            a0.y = sT[(ka + 1) * LDk + mT + l15];
            a1.x = sT[ka * LDk + mT + 32 + l15];
            a1.y = sT[(ka + 1) * LDk + mT + 32 + l15];
            bb.x = sPm[ka * LDk + nT + l15];
            bb.y = sPm[(ka + 1) * LDk + nT + l15];
            acc0 = __builtin_amdgcn_wmma_f32_16x16x4_f32(
                false, a0, false, bb, (short)0, acc0, false, false);
            acc1 = __builtin_amdgcn_wmma_f32_16x16x4_f32(
                false, a1, false, bb, (short)0, acc1, false, false);
        }
        const int bin = nT + l15;
        if (bin < BINSk) {
#pragma unroll
            for (int r = 0; r < 8; ++r) {
                const int cc = mT + r + half * 8;
                sP[cc * BINSk + bin]        = acc0[r];
                sP[(cc + 32) * BINSk + bin] = acc1[r];
            }
        }
    }
    __syncthreads();

    // ---- scatter pooled chunk to destination slots (NT stores) ----
    const int ndst = (m < Nk) ? (Nk - 1) : 1;
    for (int d = 0; d < ndst; ++d) {
        int p, slot;
        if (m < Nk) {
            const int i = m;
            const int lead = Nk - 1 - i;     // pairs (i, j>i) -> slot 0
            if (d < lead) {
                slot = 0;
                p = i * (Nk - 1) - (i * (i - 1)) / 2 + d;
            } else {                          // pairs (i2<i, i) -> slot 1
                const int i2 = d - lead;
                slot = 1;
                p = i2 * (Nk - 1) - (i2 * (i2 - 1)) / 2 + (i - i2 - 1);
            }
        } else {
            slot = 2;
            p = m - Nk;
        }
        const size_t ob = (((size_t)(b * Pk + p)) * 3 + slot) * (size_t)(Ck * BINSk)
                        + (size_t)cbase * BINSk;
        for (int i = tid; i < CCHk * BINSk; i += TPBk)
            __builtin_nontemporal_store(sP[i], &out[ob + i]);
    }
}

extern "C" void kernel_launch(void* const* d_in, const int* in_sizes, int n_in,
                              void* d_out, int out_size, void* d_ws, size_t ws_size,
                              hipStream_t stream) {
    const float* feat  = (const float*)d_in[0];   // (B, C, H, W) fp32
    const float* boxes = (const float*)d_in[1];   // (B, N, 4) fp32
    float* out = (float*)d_out;                   // (B*P, 3, C, 7, 7) fp32
    dim3 grid(Bk * Mk, Ck / CCHk);                // 1200 rois x 4 channel chunks
    roi_pair_wmma_kernel<<<grid, TPBk, 0, stream>>>(feat, boxes, out);
}